// GC_WeightNet_28690381537965
// MI455X (gfx1250) — compile-verified
//
#include <hip/hip_runtime.h>
#include <hip/hip_bf16.h>
#include <math.h>

// ---- problem constants (match reference) ----
#define BB    32
#define CIN   128
#define HH    64
#define WW    64
#define HW    4096           // 64*64
#define COUT  128
#define KDIM  1152           // CIN*3*3
#define NWPS  147456         // per-sample generated weights = COUT*KDIM
#define RC    16
#define MO    256
#define OPG   576

typedef float v2f __attribute__((ext_vector_type(2)));
typedef float v8f __attribute__((ext_vector_type(8)));

// ---------------------------------------------------------------------------
// K1: logits = x . mask_w (+b), softmax over HW -> attn[b][s]
// ---------------------------------------------------------------------------
__global__ __launch_bounds__(256) void k_attn(const float* __restrict__ x,
                                              const float* __restrict__ mw,
                                              const float* __restrict__ mb,
                                              float* __restrict__ attn) {
  const int b = blockIdx.x;
  const int tid = threadIdx.x;
  const float* xb = x + (size_t)b * CIN * HW;

  float acc[16];
#pragma unroll
  for (int j = 0; j < 16; ++j) acc[j] = 0.f;

  for (int c = 0; c < CIN; ++c) {
    const float wc = mw[c];
    const float* xc = xb + (size_t)c * HW;
#pragma unroll
    for (int j = 0; j < 16; ++j) acc[j] += xc[tid + j * 256] * wc;
  }
  const float bias = mb[0];
#pragma unroll
  for (int j = 0; j < 16; ++j) acc[j] += bias;

  __shared__ float red[256];
  float m = acc[0];
#pragma unroll
  for (int j = 1; j < 16; ++j) m = fmaxf(m, acc[j]);
  red[tid] = m;
  __syncthreads();
  for (int s = 128; s > 0; s >>= 1) {
    if (tid < s) red[tid] = fmaxf(red[tid], red[tid + s]);
    __syncthreads();
  }
  m = red[0];
  __syncthreads();
  float e[16];
  float sum = 0.f;
#pragma unroll
  for (int j = 0; j < 16; ++j) { e[j] = __expf(acc[j] - m); sum += e[j]; }
  red[tid] = sum;
  __syncthreads();
  for (int s = 128; s > 0; s >>= 1) {
    if (tid < s) red[tid] += red[tid + s];
    __syncthreads();
  }
  const float inv = 1.f / red[0];
#pragma unroll
  for (int j = 0; j < 16; ++j) attn[(size_t)b * HW + tid + j * 256] = e[j] * inv;
}

// ---------------------------------------------------------------------------
// K2: context[b][c] = sum_s x[b,c,s] * attn[b,s]
// ---------------------------------------------------------------------------
__global__ __launch_bounds__(256) void k_ctx(const float* __restrict__ x,
                                             const float* __restrict__ attn,
                                             float* __restrict__ ctx) {
  const int bc = blockIdx.x;            // b*CIN + c
  const int b = bc >> 7;
  const int tid = threadIdx.x;
  const float* xc = x + (size_t)bc * HW;
  const float* ab = attn + (size_t)b * HW;

  float s = 0.f;
#pragma unroll
  for (int j = 0; j < 16; ++j) s += xc[tid + j * 256] * ab[tid + j * 256];

  __shared__ float red[256];
  red[tid] = s;
  __syncthreads();
  for (int k = 128; k > 0; k >>= 1) {
    if (tid < k) red[tid] += red[tid + k];
    __syncthreads();
  }
  if (tid == 0) ctx[bc] = red[0];
}

// ---------------------------------------------------------------------------
// K3: t = ctx@t1^T + b; LayerNorm(16); ReLU; xw = sigmoid(t@t2^T + b)
// ---------------------------------------------------------------------------
__global__ __launch_bounds__(256) void k_mlp(const float* __restrict__ ctx,
                                             const float* __restrict__ t1w,
                                             const float* __restrict__ t1b,
                                             const float* __restrict__ lng,
                                             const float* __restrict__ lnb,
                                             const float* __restrict__ t2w,
                                             const float* __restrict__ t2b,
                                             float* __restrict__ xw) {
  __shared__ float c[BB * CIN];   // 4096
  __shared__ float t[BB * RC];    // 512
  const int tid = threadIdx.x;

  for (int i = tid; i < BB * CIN; i += 256) c[i] = ctx[i];
  __syncthreads();

  for (int idx = tid; idx < BB * RC; idx += 256) {
    const int b = idx >> 4, r = idx & 15;
    float s = t1b[r];
    for (int cc = 0; cc < CIN; ++cc) s += c[b * CIN + cc] * t1w[r * CIN + cc];
    t[idx] = s;
  }
  __syncthreads();

  if (tid < BB) {
    float mu = 0.f;
#pragma unroll
    for (int r = 0; r < RC; ++r) mu += t[tid * RC + r];
    mu *= (1.f / RC);
    float var = 0.f;
#pragma unroll
    for (int r = 0; r < RC; ++r) {
      const float d = t[tid * RC + r] - mu;
      var += d * d;
    }
    var *= (1.f / RC);
    const float inv = rsqrtf(var + 1e-5f);
#pragma unroll
    for (int r = 0; r < RC; ++r) {
      const float v = (t[tid * RC + r] - mu) * inv * lng[r] + lnb[r];
      t[tid * RC + r] = fmaxf(v, 0.f);
    }
  }
  __syncthreads();

  for (int idx = tid; idx < BB * MO; idx += 256) {
    const int b = idx >> 8, o = idx & 255;
    float s = t2b[o];
#pragma unroll
    for (int r = 0; r < RC; ++r) s += t[b * RC + r] * t2w[o * RC + r];
    xw[idx] = 1.f / (1.f + __expf(-s));
  }
}

// ---------------------------------------------------------------------------
// K4: generate per-sample conv weights in TAP-MAJOR, K-PAIRED layout:
//   new k-index  k = tap*128 + ci      (tap = kh*3 + kw)
//   storage      wg[b][kp][co][2]      (kp = k>>1, last dim = k&1)
// so the conv's A fragment is one aligned, lane-coalesced b64 load and the
// K-loop needs no div/mod. Original flat f = co*1152 + ci*9 + tap, grp=f/576.
// ---------------------------------------------------------------------------
__global__ __launch_bounds__(256) void k_wgen(const float* __restrict__ fcw,
                                              const float* __restrict__ fcb,
                                              const float* __restrict__ xw,
                                              float* __restrict__ wg) {
  const size_t g = (size_t)blockIdx.x * 256 + threadIdx.x;   // exactly BB*NWPS
  const int b = (int)(g / NWPS);
  const int rem = (int)(g % NWPS);
  const int kp = rem >> 8;            // k-pair index
  const int t2 = rem & 255;
  const int co = t2 >> 1;
  const int j  = t2 & 1;
  const int k  = kp * 2 + j;          // tap*128 + ci
  const int tap = k >> 7;
  const int ci  = k & 127;
  const int f = co * KDIM + ci * 9 + tap;
  const int grp = f / OPG;
  wg[g] = fcw[f] * xw[b * MO + grp] + fcb[f];
}

// ---------------------------------------------------------------------------
// K5: per-sample 3x3 conv as 9 accumulating GEMMs (one per tap), each K=128,
// using V_WMMA_F32_16X16X4_F32 (f32 keeps exact precision; op is mem-bound).
//
// Block = 256 threads (8 waves), covers ALL 128 co x 2 rows x 32 cols (N=64).
// Each wave holds 4 accumulators (row r in {0,1} x col-subtile c in {0,1})
// and reuses one A fragment across 4 WMMAs -> 4x A-traffic reduction.
// ci is processed in 2 phases of 64 (LDS tile 36 KiB).
//
// LDS tile is CI-PAIR-INTERLEAVED: xs[cipair][winrow][col][2] so a B fragment
// {K=ci, K=ci+1} at the same (row, col) is one aligned ds_load_b64 straight
// into an adjacent VGPR pair (no v_mov fragment fixups).
//
// ISA layouts (7.12.2):
//   A 16x4 f32: lanes<16 -> {K,K+1}, lanes>=16 -> {K+2,K+3}, M = lane%16
//   B 4x16 f32: mirrored; C/D: VGPR j -> M=j (lanes<16) / M=j+8 (lanes>=16)
// Full 256-thread blocks, uniform control flow around WMMA => EXEC all-1s.
// ---------------------------------------------------------------------------
__global__ __launch_bounds__(256) void k_conv(const float* __restrict__ x,
                                              const float* __restrict__ wg,
                                              float* __restrict__ out) {
  const int x0 = blockIdx.x * 32;       // 0 or 32
  const int y0 = blockIdx.y * 2;        // 0,2,..,62
  const int b  = blockIdx.z;            // 0..31
  const int tid = threadIdx.x;

  // [cipair(32)][winrow(4)][col(36)][pair(2)] = 36 KiB
  __shared__ float xs[32 * 4 * 36 * 2];

  const int wave = tid >> 5;
  const int lane = tid & 31;
  const int n = lane & 15;
  const int hi = lane >> 4;
  const int co_base = wave * 16;

  const float* xb  = x + (size_t)b * CIN * HW;
  const float* wkb = wg + (size_t)b * NWPS;   // [kp][co][2] pair layout

  v8f a00 = {0.f,0.f,0.f,0.f,0.f,0.f,0.f,0.f};
  v8f a01 = a00, a10 = a00, a11 = a00;

  for (int ph = 0; ph < 2; ++ph) {
    if (ph) __syncthreads();            // all waves done reading prev tile
    // ---- stage halo tile: ci in [ph*64, ph*64+64), rows y0-1..y0+2 ----
    // global reads coalesced (col fastest); LDS writes pair-interleaved
    for (int e = tid; e < 64 * 4 * 36; e += 256) {
      const int ci = e / 144;
      const int rem = e - ci * 144;
      const int kh = rem / 36;
      const int col = rem - kh * 36;
      const int gy = y0 + kh - 1;
      const int gx = x0 + col - 1;
      float v = 0.f;
      if (gy >= 0 && gy < HH && gx >= 0 && gx < WW)
        v = xb[(size_t)(ph * 64 + ci) * HW + gy * WW + gx];
      xs[(ci >> 1) * 288 + kh * 72 + col * 2 + (ci & 1)] = v;
    }
    __syncthreads();

    // ---- 9 taps x 16 K-steps of 4 ----
    for (int tap = 0; tap < 9; ++tap) {
      const int kh = tap / 3;           // uniform scalar
      const int kw = tap - kh * 3;
      const float* abase = wkb + ((size_t)(tap * 64 + ph * 32 + hi) * 256)
                               + (co_base + n) * 2;
      if (tap < 8) __builtin_prefetch(abase + 64 * 256, 0, 3);
      // window col for subtile0 = n + kw  ->  float offset = (n+kw)*2
      const int bbase = kh * 72 + (kw + n) * 2;

#pragma unroll 4
      for (int cb = 0; cb < 64; cb += 4) {
        // A: rows co_base..+15, K = k0,k0+1 (one coalesced b64 per lane)
        const v2f a = *(const v2f*)(abase + (cb >> 1) * 256);

        // B: ci-pair (cb/2 + hi), each fragment one aligned ds_load_b64
        const int ib = (cb / 2 + hi) * 288 + bbase;
        const v2f b00 = *(const v2f*)&xs[ib];            // row0, cols n
        const v2f b01 = *(const v2f*)&xs[ib + 32];       // row0, cols n+16
        const v2f b10 = *(const v2f*)&xs[ib + 72];       // row1, cols n
        const v2f b11 = *(const v2f*)&xs[ib + 104];      // row1, cols n+16

        a00 = __builtin_amdgcn_wmma_f32_16x16x4_f32(false, a, false, b00,
                                                    (short)0, a00, false, false);
        a01 = __builtin_amdgcn_wmma_f32_16x16x4_f32(false, a, false, b01,
                                                    (short)0, a01, false, false);
        a10 = __builtin_amdgcn_wmma_f32_16x16x4_f32(false, a, false, b10,
                                                    (short)0, a10, false, false);
        a11 = __builtin_amdgcn_wmma_f32_16x16x4_f32(false, a, false, b11,
                                                    (short)0, a11, false, false);
      }
    }
  }

  // ---- store: VGPR j -> M = j (lanes 0-15) / M = j+8 (lanes 16-31) ----
  const int row0 = co_base + (hi ? 8 : 0);
  float* ob = out + (size_t)b * COUT * HW + y0 * WW + x0 + n;
#pragma unroll
  for (int j = 0; j < 8; ++j) {
    float* orow = ob + (size_t)(row0 + j) * HW;
    orow[0]       = a00[j];
    orow[16]      = a01[j];
    orow[WW]      = a10[j];
    orow[WW + 16] = a11[j];
  }
}

// ---------------------------------------------------------------------------
extern "C" void kernel_launch(void* const* d_in, const int* in_sizes, int n_in,
                              void* d_out, int out_size, void* d_ws, size_t ws_size,
                              hipStream_t stream) {
  const float* x      = (const float*)d_in[0];
  const float* mask_w = (const float*)d_in[1];
  const float* mask_b = (const float*)d_in[2];
  const float* t1w    = (const float*)d_in[3];
  const float* t1b    = (const float*)d_in[4];
  const float* lng    = (const float*)d_in[5];
  const float* lnb    = (const float*)d_in[6];
  const float* t2w    = (const float*)d_in[7];
  const float* t2b    = (const float*)d_in[8];
  const float* fcw    = (const float*)d_in[9];
  const float* fcb    = (const float*)d_in[10];
  float* out = (float*)d_out;

  // workspace layout (floats): attn | ctx | xw | wgen  (~18.6 MiB total)
  float* ws   = (float*)d_ws;
  float* attn = ws;                       // 32*4096      = 131072
  float* ctx  = ws + 131072;              // 32*128       = 4096
  float* xw   = ws + 135168;              // 32*256       = 8192
  float* wgen = ws + 143360;              // 32*147456    = 4718592

  k_attn<<<BB, 256, 0, stream>>>(x, mask_w, mask_b, attn);
  k_ctx<<<BB * CIN, 256, 0, stream>>>(x, attn, ctx);
  k_mlp<<<1, 256, 0, stream>>>(ctx, t1w, t1b, lng, lnb, t2w, t2b, xw);
  k_wgen<<<(BB * NWPS) / 256, 256, 0, stream>>>(fcw, fcb, xw, wgen);
  dim3 cg(2, HH / 2, BB);
  k_conv<<<cg, 256, 0, stream>>>(x, wgen, out);
}